// QualityFeatureExtractor_85409719649039
// MI455X (gfx1250) — compile-verified
//
#include <hip/hip_runtime.h>
#include <cmath>

typedef __attribute__((ext_vector_type(2))) float v2f;
typedef __attribute__((ext_vector_type(4))) float v4f;
typedef __attribute__((ext_vector_type(8))) float v8f;

#define NROWS    262144
#define EMBED    768
#define HID      64
#define LN_EPS   1e-5f
#define LDS_PITCH 68                       // 68 % 64 == 4 -> conflict-free B reads
#define LDS_BYTES (EMBED * LDS_PITCH * 4)  // 208896 B <= 320KB WGP LDS

// One wave handles a 16-row strip x full 768 cols.
// Block = 256 threads = 8 waves = 128 rows. Grid = 262144/128 = 2048.
__global__ __launch_bounds__(256)
void qfe_wmma_kernel(const float* __restrict__ qs,        // (B,1)
                     const float* __restrict__ level_emb, // (5,768)
                     const float* __restrict__ w1,        // (64,1)
                     const float* __restrict__ b1,        // (64,)
                     const float* __restrict__ gma,       // (64,)
                     const float* __restrict__ bta,       // (64,)
                     const float* __restrict__ w2,        // (768,64) row-major
                     const float* __restrict__ b2,        // (768,)
                     float* __restrict__ out)             // (B,768)
{
    extern __shared__ float sB[];          // w2 staged, row pitch 68 floats

    // ---------------- stage w2 -> LDS (once per block, shared by 8 waves) -----
    // 768 rows x 16 chunks of 16B; 256 threads x 48 iters; coalesced b128 loads.
    for (int m = threadIdx.x; m < EMBED * 16; m += 256) {
        const int r = m >> 4;
        const int c = (m & 15) << 2;
        v4f v = *(const v4f*)(w2 + (size_t)r * HID + c);
        *(v4f*)(&sB[r * LDS_PITCH + c]) = v;
    }

    const int lane    = threadIdx.x & 31;
    const int wave    = threadIdx.x >> 5;
    const int strip   = blockIdx.x * 8 + wave;   // 16-row strip
    const int row0    = strip * 16;
    const int halfsel = lane >> 4;               // 0: lanes 0-15, 1: lanes 16-31
    const int lmod    = lane & 15;
    const int koff    = halfsel * 2;             // A/B K sub-offset per ISA layout

    // ---------------- build A operand: g(s) for row (row0+lmod) ----------------
    // A 16x4 f32 layout: lane holds M = lane%16; VGPR0 = K(0|2), VGPR1 = K(1|3).
    const float s = qs[row0 + lmod];

    float sum = 0.f, sumsq = 0.f;
    #pragma unroll
    for (int j = 0; j < HID; ++j) {
        float h = fmaf(s, w1[j], b1[j]);        // w1/b1 wave-uniform -> s_load
        sum   += h;
        sumsq += h * h;
    }
    const float mu   = sum * (1.0f / HID);
    const float var  = sumsq * (1.0f / HID) - mu * mu;
    const float rstd = rsqrtf(var + LN_EPS);

    v2f a[16];
    #pragma unroll
    for (int j = 0; j < HID; ++j) {
        float h  = fmaf(s, w1[j], b1[j]);
        float hn = (h - mu) * rstd * gma[j] + bta[j];
        // exact GELU: 0.5*x*(1+erf(x/sqrt(2)))
        float g  = 0.5f * hn * (1.0f + erff(hn * 0.70710678118654752f));
        const int jm = j & 3, ji = j >> 2;
        if (jm == koff)          a[ji].x = g;   // lane-half picks its K elements
        else if (jm == koff + 1) a[ji].y = g;
    }

    // ---------------- per-lane embedding lerp params for its 8 C-rows ----------
    // C/D layout: VGPR r -> row (row0 + r + 8*halfsel), col (n0 + lmod).
    float wlo[8], whi[8];
    int   lidx[8];
    #pragma unroll
    for (int r = 0; r < 8; ++r) {
        float sr    = qs[row0 + r + halfsel * 8];
        float lower = fminf(fmaxf(floorf(sr - 1.0f), 0.0f), 3.0f);
        float alpha = sr - (lower + 1.0f);
        lidx[r] = (int)lower;
        whi[r]  = alpha;
        wlo[r]  = 1.0f - alpha;
    }

    __syncthreads();   // w2 staged

    // per-lane base into staged B: row lmod, K sub-offset koff
    const float* lB = sB + lmod * LDS_PITCH + koff;

    // ---------------- GEMM: 48 col tiles, 2 at a time (independent chains) ----
    for (int t = 0; t < 48; t += 2) {
        const int n0 = t * 16;
        const float* p0 = lB + (size_t)n0 * LDS_PITCH;
        const float* p1 = p0 + (size_t)16 * LDS_PITCH;

        v8f c0 = {}, c1 = {};
        #pragma unroll
        for (int k = 0; k < 16; ++k) {
            v2f bv0 = *(const v2f*)(p0 + 4 * k);   // ds_load_b64, bank-conflict-free
            v2f bv1 = *(const v2f*)(p1 + 4 * k);
            c0 = __builtin_amdgcn_wmma_f32_16x16x4_f32(false, a[k], false, bv0,
                                                       (short)0, c0, false, false);
            c1 = __builtin_amdgcn_wmma_f32_16x16x4_f32(false, a[k], false, bv1,
                                                       (short)0, c1, false, false);
        }

        // ---------------- epilogue: + b2 + stop-grad embedding lerp ------------
        #pragma unroll
        for (int half = 0; half < 2; ++half) {
            const int n = n0 + half * 16 + lmod;
            const float bias = b2[n];
            const v8f& c = half ? c1 : c0;
            #pragma unroll
            for (int r = 0; r < 8; ++r) {
                const int li = lidx[r];
                float e = wlo[r] * level_emb[li * EMBED + n]
                        + whi[r] * level_emb[(li + 1) * EMBED + n];
                out[(size_t)(row0 + r + halfsel * 8) * EMBED + n] = c[r] + bias + e;
            }
        }
    }
}

extern "C" void kernel_launch(void* const* d_in, const int* in_sizes, int n_in,
                              void* d_out, int out_size, void* d_ws, size_t ws_size,
                              hipStream_t stream) {
    (void)in_sizes; (void)n_in; (void)out_size; (void)d_ws; (void)ws_size;
    const float* qs        = (const float*)d_in[0];
    const float* level_emb = (const float*)d_in[1];
    const float* w1        = (const float*)d_in[2];
    const float* b1        = (const float*)d_in[3];
    const float* gma       = (const float*)d_in[4];
    const float* bta       = (const float*)d_in[5];
    const float* w2        = (const float*)d_in[6];
    const float* b2        = (const float*)d_in[7];
    float* out             = (float*)d_out;

    dim3 grid(NROWS / 128);   // 8 waves/block * 16 rows/wave
    dim3 block(256);
    qfe_wmma_kernel<<<grid, block, LDS_BYTES, stream>>>(qs, level_emb, w1, b1,
                                                        gma, bta, w2, b2, out);
}